// FT_model_32624571580944
// MI455X (gfx1250) — compile-verified
//
#include <hip/hip_runtime.h>
#include <hip/hip_bf16.h>

typedef __attribute__((ext_vector_type(2))) float v2f;
typedef __attribute__((ext_vector_type(8))) float v8f;

// ---------------------------------------------------------------------------
// Pack a 16xK row-major matrix into V_WMMA_F32_16X16X4_F32 A-fragment order.
// Chunk c covers K=[4c,4c+4). Lane L (0..31): m = L&15,
//   v0 = A[m][4c + 2*(L>=16)], v1 = A[m][4c + 2*(L>=16) + 1]
// Stored as float2 at out[c*32 + L] -> GEMM A-loads are coalesced b64.
// ---------------------------------------------------------------------------
__global__ void pack_a(const float* __restrict__ A, float2* __restrict__ out, int K)
{
    int id = blockIdx.x * blockDim.x + threadIdx.x;
    int total = (K >> 2) * 32;
    if (id >= total) return;
    int c = id >> 5;
    int L = id & 31;
    int m = L & 15;
    int k = (c << 2) + ((L >> 4) << 1);
    const float* a = A + (size_t)m * K + k;
    out[id] = make_float2(a[0], a[1]);
}

__device__ __forceinline__ float fast_sigmoid(float x)
{
    return __builtin_amdgcn_rcpf(1.0f + __expf(-x));
}

// ---------------------------------------------------------------------------
// D(16xN) = A(16xK) @ W(KxN) [+ bias] [sigmoid], one 16-col tile per wave.
// N is a compile-time constant so the 16*N-byte k-step strides fold into the
// 24-bit instruction immediate offsets (no per-step 64-bit address VALU).
// W streamed with non-temporal loads to keep L2 for activations.
// ---------------------------------------------------------------------------
template<int N, bool BIAS, bool ACT>
__global__ void gemm16_full(const float2* __restrict__ Ap,
                            const float* __restrict__ W,
                            const float* __restrict__ bias,
                            float* __restrict__ out,
                            int K)
{
    const int lane = threadIdx.x & 31;
    const int wave = (int)((blockIdx.x * blockDim.x + threadIdx.x) >> 5);
    const int n0   = wave << 4;
    const int s0   = (lane >> 4) << 1;   // K sub-row for this lane half (0 or 2)
    const int nc   = lane & 15;
    const int chunks = K >> 2;

    const float2* ap = Ap + lane;
    const float*  wp = W + (size_t)s0 * N + (size_t)n0 + nc;

    v8f acc = {0.f, 0.f, 0.f, 0.f, 0.f, 0.f, 0.f, 0.f};
#pragma unroll 8
    for (int c = 0; c < chunks; ++c) {
        float2 a2 = ap[(size_t)c << 5];
        v2f a; a.x = a2.x; a.y = a2.y;
        const float* w = wp + (size_t)c * (size_t)(4 * N);
        v2f b; b.x = __builtin_nontemporal_load(w);
               b.y = __builtin_nontemporal_load(w + N);
        acc = __builtin_amdgcn_wmma_f32_16x16x4_f32(false, a, false, b,
                                                    (short)0, acc, false, false);
    }

    const int col = n0 + nc;
    const float bv = BIAS ? bias[col] : 0.0f;
    const int m0 = (lane >> 4) << 3;     // C layout: VGPR r -> M = r + 8*(lane>=16)
#pragma unroll
    for (int r = 0; r < 8; ++r) {
        float v = acc[r] + bv;
        if (ACT) v = fast_sigmoid(v);
        out[(size_t)(m0 + r) * N + col] = v;
    }
}

// ---------------------------------------------------------------------------
// K-split version: wave = (slab, tile); accumulates CPS chunks starting at
// slab*CPS and writes the raw 16x16 partial tile in lane-fragment order.
// ---------------------------------------------------------------------------
template<int N, int NTILES, int CPS>
__global__ void gemm16_part(const float2* __restrict__ Ap,
                            const float* __restrict__ W,
                            float* __restrict__ part)
{
    const int lane = threadIdx.x & 31;
    const int wid  = (int)((blockIdx.x * blockDim.x + threadIdx.x) >> 5);
    const int tile = wid & (NTILES - 1);
    const int slab = wid / NTILES;
    const int n0   = tile << 4;
    const int c0   = slab * CPS;
    const int s0   = (lane >> 4) << 1;
    const int nc   = lane & 15;

    const float2* ap = Ap + ((size_t)c0 << 5) + lane;
    const float*  wp = W + ((size_t)(4 * c0) + s0) * N + (size_t)n0 + nc;

    v8f acc = {0.f, 0.f, 0.f, 0.f, 0.f, 0.f, 0.f, 0.f};
#pragma unroll 8
    for (int c = 0; c < CPS; ++c) {
        float2 a2 = ap[(size_t)c << 5];
        v2f a; a.x = a2.x; a.y = a2.y;
        const float* w = wp + (size_t)c * (size_t)(4 * N);
        v2f b; b.x = __builtin_nontemporal_load(w);
               b.y = __builtin_nontemporal_load(w + N);
        acc = __builtin_amdgcn_wmma_f32_16x16x4_f32(false, a, false, b,
                                                    (short)0, acc, false, false);
    }

    float* p = part + (((size_t)slab * NTILES + tile) << 8) + lane * 8;
    float4 lo = make_float4(acc[0], acc[1], acc[2], acc[3]);
    float4 hi = make_float4(acc[4], acc[5], acc[6], acc[7]);
    *(float4*)(p)     = lo;
    *(float4*)(p + 4) = hi;
}

// Sum 32 K-slab partials (stored in fragment order) + bias -> z row-major 16x2048
__global__ void zreduce(const float* __restrict__ part,
                        const float* __restrict__ b2,
                        float* __restrict__ z)
{
    int tid  = blockIdx.x * blockDim.x + threadIdx.x;   // 0..32767
    int m    = tid >> 11;
    int col  = tid & 2047;
    int tile = col >> 4;
    int n    = col & 15;
    int lane = n + ((m >= 8) ? 16 : 0);
    int r    = m & 7;
    float s  = b2[col];
    const float* p = part + ((size_t)tile << 8) + lane * 8 + r;
#pragma unroll 8
    for (int sl = 0; sl < 32; ++sl)
        s += p[(size_t)sl * 128 * 256];
    z[tid] = s;   // m*2048 + col == tid
}

// ---------------------------------------------------------------------------
// Decode: out[b,t] = (1/1024) * sum_k ( yr[b,k]*cos(2*pi*t*k/1024)
//                                     - yi[b,k]*sin(2*pi*t*k/1024) )
// (identical to amp*cos(arg+ph) with amp=r/L, ph=atan2). 4-way K-split.
// v_sin/v_cos take 2*pi-normalized input: frac = (t*k mod 1024)/1024.
// ---------------------------------------------------------------------------
__global__ void decode_part(const float* __restrict__ z, float* __restrict__ dpart)
{
    __shared__ float syr[256];
    __shared__ float syi[256];
    const int b    = blockIdx.x >> 2;
    const int slab = blockIdx.x & 3;
    const int k0   = slab << 8;
    const int t    = threadIdx.x;          // 0..1023
    if (t < 256)       syr[t]        = z[b * 2048 + k0 + t];
    else if (t < 512)  syi[t - 256]  = z[b * 2048 + 1024 + k0 + (t - 256)];
    __syncthreads();
    float sum = 0.0f;
#pragma unroll 4
    for (int kk = 0; kk < 256; ++kk) {
        int   k    = k0 + kk;
        int   tk   = (t * k) & 1023;
        float frac = (float)tk * (1.0f / 1024.0f);
        float c = __builtin_amdgcn_cosf(frac);
        float s = __builtin_amdgcn_sinf(frac);
        sum += syr[kk] * c - syi[kk] * s;
    }
    dpart[((size_t)blockIdx.x << 10) + t] = sum;
}

__global__ void decode_reduce(const float* __restrict__ dpart, float* __restrict__ out)
{
    int tid  = blockIdx.x * blockDim.x + threadIdx.x;   // 0..16383
    int base = ((tid >> 10) << 12) + (tid & 1023);
    float s = dpart[base] + dpart[base + 1024] + dpart[base + 2048] + dpart[base + 3072];
    out[tid] = s * (1.0f / 1024.0f);
}

// ---------------------------------------------------------------------------
extern "C" void kernel_launch(void* const* d_in, const int* in_sizes, int n_in,
                              void* d_out, int out_size, void* d_ws, size_t ws_size,
                              hipStream_t stream)
{
    const float* X  = (const float*)d_in[0];   // 16 x 1024
    const float* Fc = (const float*)d_in[1];   // 1024 x 2048
    const float* W1 = (const float*)d_in[2];   // 2048 x 51200
    const float* b1 = (const float*)d_in[3];   // 51200
    const float* W2 = (const float*)d_in[4];   // 51200 x 2048
    const float* b2 = (const float*)d_in[5];   // 2048
    float* out = (float*)d_out;                // 16 x 1024

    char* ws = (char*)d_ws;
    size_t off = 0;
    auto carve = [&](size_t bytes) -> char* {
        char* p = ws + off;
        off = (off + bytes + 255) & ~(size_t)255;
        return p;
    };
    float2* Xp    = (float2*)carve((size_t)8192   * sizeof(float2));  // packed X (K=1024)
    float*  y     = (float*) carve((size_t)16 * 2048  * sizeof(float));
    float2* yA    = (float2*)carve((size_t)16384  * sizeof(float2));  // packed y (K=2048)
    float*  h     = (float*) carve((size_t)16 * 51200 * sizeof(float));
    float2* hA    = (float2*)carve((size_t)409600 * sizeof(float2));  // packed h (K=51200)
    float*  part  = (float*) carve((size_t)32 * 128 * 256 * sizeof(float));
    float*  z     = (float*) carve((size_t)16 * 2048  * sizeof(float));
    float*  dpart = (float*) carve((size_t)16 * 4 * 1024 * sizeof(float));
    (void)ws_size; (void)in_sizes; (void)n_in; (void)out_size;

    // Stage 1: y = X @ Fc             (M=16, K=1024, N=2048 -> 128 tiles)
    pack_a<<<32, 256, 0, stream>>>(X, Xp, 1024);
    gemm16_full<2048, false, false><<<16, 256, 0, stream>>>(Xp, Fc, nullptr, y, 1024);

    // Stage 2: h = sigmoid(y @ W1 + b1)   (K=2048, N=51200 -> 3200 tiles)
    pack_a<<<64, 256, 0, stream>>>(y, yA, 2048);
    gemm16_full<51200, true, true><<<400, 256, 0, stream>>>(yA, W1, b1, h, 2048);

    // Stage 3: z = h @ W2 + b2       (K=51200 split 32 ways, N=2048 -> 128 tiles)
    pack_a<<<1600, 256, 0, stream>>>(h, hA, 51200);
    gemm16_part<2048, 128, 400><<<512, 256, 0, stream>>>(hA, W2, part);
    zreduce<<<128, 256, 0, stream>>>(part, b2, z);

    // Stage 4: inverse-DFT decode
    decode_part<<<64, 1024, 0, stream>>>(z, dpart);
    decode_reduce<<<64, 256, 0, stream>>>(dpart, out);
}